// MoEsparseRoutingForClassification_13030930776373
// MI455X (gfx1250) — compile-verified
//
#include <hip/hip_runtime.h>
#include <math.h>

// Problem constants (match reference)
#define B_  64
#define S_  128
#define H_  1024
#define E_  8
#define L_  2
#define K_  (E_ * H_)      // 8192 folded reduction dim
#define KSPLIT 8           // one expert per K-slice

typedef float v2f __attribute__((ext_vector_type(2)));
typedef float v4f __attribute__((ext_vector_type(4)));
typedef float v8f __attribute__((ext_vector_type(8)));

// ---------------------------------------------------------------------------
// K0: A[b, e*H + i] = gates[b,e] * X[b,0,i]          (64 x 8192 fp32, 2 MB)
// ---------------------------------------------------------------------------
__global__ void build_A_kernel(const float* __restrict__ X,
                               const float* __restrict__ gates,
                               float* __restrict__ A) {
    const int b = blockIdx.x;                       // 0..63
    const v4f* xrow = (const v4f*)(X + (size_t)b * S_ * H_);   // X[b,0,:]
    v4f*       arow = (v4f*)(A + (size_t)b * K_);
    // 2048 float4 per row; e = (t*4)>>10 = t>>8 (constant within a float4)
    for (int t = threadIdx.x; t < K_ / 4; t += blockDim.x) {
        const float g = gates[b * E_ + (t >> 8)];
        v4f v = xrow[t & ((H_ / 4) - 1)];
        v.x *= g; v.y *= g; v.z *= g; v.w *= g;
        arow[t] = v;
    }
}

// ---------------------------------------------------------------------------
// K1: Ypart[ks][m][o] = A[m, ks*1024 : (ks+1)*1024] @ dense_w[ks][o][:]
//     grid = (256 N-tiles, 8 K-slices), block = 128 threads = 4 waves.
//     Wave w computes the 16x16 tile for M rows [w*16, w*16+16).
//     v_wmma_f32_16x16x4_f32, fp32 end to end.
// ---------------------------------------------------------------------------
__global__ __launch_bounds__(128)
void gemm_wmma_kernel(const float* __restrict__ A,        // 64 x 8192
                      const float* __restrict__ dense_w,  // E x H x H
                      float* __restrict__ Ypart) {        // KSPLIT x 64 x 1024
    const int ntile  = blockIdx.x;        // 0..255   (16 output cols each)
    const int kslice = blockIdx.y;        // 0..7    == expert index e
    const int wave   = threadIdx.x >> 5;  // 0..3    == M tile
    const int lane   = threadIdx.x & 31;
    const int col    = lane & 15;         // N (B frag) / M (A frag) within tile
    const int half   = lane >> 4;         // K sub-offset selector

    // A fragment: lane holds M = wave*16 + col, K = k0 + 2*half (+1)
    const float* __restrict__ Abase =
        A + (size_t)(wave * 16 + col) * K_ + (size_t)kslice * H_ + 2 * half;
    // B fragment: lane holds N = ntile*16 + col, K = k0 + 2*half (+1).
    // B[(e,i), o] = dense_w[e*H*H + o*H + i] -> column o is contiguous in i.
    const float* __restrict__ Bbase =
        dense_w + (size_t)kslice * H_ * H_ + (size_t)(ntile * 16 + col) * H_ + 2 * half;

    v8f acc = {};
#pragma unroll 8
    for (int i0 = 0; i0 < H_; i0 += 4) {
        v2f a = *(const v2f*)(Abase + i0);
        v2f b = *(const v2f*)(Bbase + i0);
        // (neg_a, A, neg_b, B, c_mod, C, reuse_a, reuse_b)
        acc = __builtin_amdgcn_wmma_f32_16x16x4_f32(
            false, a, false, b, (short)0, acc, false, false);
    }

    // C/D layout: VGPR r -> (lanes 0-15: M=r, lanes 16-31: M=r+8), N = col
    float* __restrict__ yp = Ypart
        + ((size_t)kslice * B_ + (size_t)(wave * 16 + 8 * half)) * H_
        + ntile * 16 + col;
#pragma unroll
    for (int r = 0; r < 8; ++r)
        yp[(size_t)r * H_] = acc[r];
}

// ---------------------------------------------------------------------------
// K2: x[b,o] = tanh( sum_ks Ypart[ks][b][o] + sum_e gates[b,e]*dense_b[e,o] )
// ---------------------------------------------------------------------------
__global__ void reduce_bias_tanh_kernel(const float* __restrict__ Ypart,
                                        const float* __restrict__ gates,
                                        const float* __restrict__ dense_b,
                                        float* __restrict__ Xact) {
    const int idx = blockIdx.x * blockDim.x + threadIdx.x;  // 0..65535
    const int b = idx >> 10;
    const int o = idx & (H_ - 1);
    float s = 0.f;
#pragma unroll
    for (int ks = 0; ks < KSPLIT; ++ks)
        s += Ypart[((size_t)ks * B_ + b) * H_ + o];
#pragma unroll
    for (int e = 0; e < E_; ++e)
        s += gates[b * E_ + e] * dense_b[e * H_ + o];
    Xact[idx] = tanhf(s);
}

// ---------------------------------------------------------------------------
// K3: out[b,l] = sum_e gates[b,e] * ( x[b,:] . out_w[e,l,:] + out_b[e,l] )
//     one block per b; 16 waves, one per (e,l) dot product.
// ---------------------------------------------------------------------------
__global__ __launch_bounds__(512)
void head_kernel(const float* __restrict__ Xact,
                 const float* __restrict__ gates,
                 const float* __restrict__ out_w,
                 const float* __restrict__ out_b,
                 float* __restrict__ out) {
    const int b    = blockIdx.x;
    const int wave = threadIdx.x >> 5;   // 0..15  -> (e = wave>>1, l = wave&1)
    const int lane = threadIdx.x & 31;

    const float* __restrict__ xr = Xact + (size_t)b * H_;
    const float* __restrict__ wr = out_w + (size_t)wave * H_;   // out_w[e][l][:]

    float p = 0.f;
#pragma unroll 4
    for (int i = lane; i < H_; i += 32)
        p += xr[i] * wr[i];
#pragma unroll
    for (int off = 16; off > 0; off >>= 1)
        p += __shfl_down(p, off, 32);

    __shared__ float t[E_ * L_];
    if (lane == 0) t[wave] = p;
    __syncthreads();

    if (threadIdx.x < L_) {
        const int l = threadIdx.x;
        float s = 0.f;
#pragma unroll
        for (int e = 0; e < E_; ++e)
            s += gates[b * E_ + e] * (t[e * L_ + l] + out_b[e * L_ + l]);
        out[b * L_ + l] = s;
    }
}

// ---------------------------------------------------------------------------
// Launch
// ---------------------------------------------------------------------------
extern "C" void kernel_launch(void* const* d_in, const int* in_sizes, int n_in,
                              void* d_out, int out_size, void* d_ws, size_t ws_size,
                              hipStream_t stream) {
    const float* X       = (const float*)d_in[0];
    const float* gates   = (const float*)d_in[1];
    const float* dense_w = (const float*)d_in[2];
    const float* dense_b = (const float*)d_in[3];
    const float* out_w   = (const float*)d_in[4];
    const float* out_b   = (const float*)d_in[5];
    float* out = (float*)d_out;

    char* ws = (char*)d_ws;
    float* A     = (float*)(ws);                                   // 2 MB
    float* Ypart = (float*)(ws + (size_t)B_ * K_ * 4);             // 2 MB
    float* Xact  = (float*)(ws + (size_t)B_ * K_ * 4
                               + (size_t)KSPLIT * B_ * H_ * 4);    // 256 KB

    build_A_kernel<<<B_, 256, 0, stream>>>(X, gates, A);

    dim3 g1(B_ * H_ / (16 * 16), KSPLIT);   // 256 N-tiles x 8 K-slices
    gemm_wmma_kernel<<<g1, 128, 0, stream>>>(A, dense_w, Ypart);

    reduce_bias_tanh_kernel<<<(B_ * H_) / 256, 256, 0, stream>>>(
        Ypart, gates, dense_b, Xact);

    head_kernel<<<B_, 512, 0, stream>>>(Xact, gates, out_w, out_b, out);
}